// UnrolledLSTMCellModel_2783138808196
// MI455X (gfx1250) — compile-verified
//
#include <hip/hip_runtime.h>
#include <hip/hip_bf16.h>

typedef float v2f __attribute__((ext_vector_type(2)));
typedef float v8f __attribute__((ext_vector_type(8)));
typedef float f2  __attribute__((ext_vector_type(2)));

#define TT   100
#define BB   16384
#define FF   64
#define G4H  20     // 4*H
#define HH   5

// ---------------------------------------------------------------------------
// Kernel 1: XZ[t][b][0:20] = x[b][t][:] @ W1 + b1   (exact f32 via WMMA)
// M dimension indexed as m = t*B + b so XZ comes out in [T][B][20] layout,
// making kernel 2's per-step reads contiguous per wave.
// W1 fragments (64x32, zero-padded cols 20..31) live in VGPRs for the whole
// kernel; per M-tile we do 16 chained V_WMMA_F32_16X16X4_F32 per N-tile.
// ---------------------------------------------------------------------------
__global__ __launch_bounds__(256) void xw1_wmma(
    const float* __restrict__ x, const float* __restrict__ W1,
    const float* __restrict__ b1, float* __restrict__ xz)
{
  const int lane = threadIdx.x & 31;
  const int half = lane >> 4;     // 0: lanes 0-15, 1: lanes 16-31
  const int lm   = lane & 15;
  const int wid  = (blockIdx.x * blockDim.x + threadIdx.x) >> 5;
  const int nw   = (gridDim.x * blockDim.x) >> 5;

  // B-matrix fragments: B is 4(K) x 16(N); VGPR r, lane l -> K = r + 2*half, N = lm.
  v2f bf0[16], bf1[16];
  #pragma unroll
  for (int s = 0; s < 16; ++s) {
    const int k0 = 4 * s + 2 * half;
    v2f t0 = { W1[k0 * G4H + lm], W1[(k0 + 1) * G4H + lm] };   // cols 0..15 always valid
    bf0[s] = t0;
    float e0 = 0.0f, e1 = 0.0f;
    const int col1 = 16 + lm;
    if (col1 < G4H) { e0 = W1[k0 * G4H + col1]; e1 = W1[(k0 + 1) * G4H + col1]; }
    v2f t1 = { e0, e1 };
    bf1[s] = t1;
  }
  const float bias0 = b1[lm];                                  // C init = bias row-broadcast
  const float bias1 = (16 + lm < G4H) ? b1[16 + lm] : 0.0f;

  const int NT = (TT * BB) / 16;                               // 102400 M-tiles
  for (int mt = wid; mt < NT; mt += nw) {
    const long m0 = (long)mt * 16;
    const int  t  = (int)(m0 >> 14);                           // B = 2^14
    const int  b0 = (int)(m0 & (BB - 1));                      // multiple of 16
    const float* xrow = x + ((long)(b0 + lm) * TT + t) * FF;   // A row for this lane

    v8f acc0 = {bias0, bias0, bias0, bias0, bias0, bias0, bias0, bias0};
    v8f acc1 = {bias1, bias1, bias1, bias1, bias1, bias1, bias1, bias1};

    #pragma unroll
    for (int s = 0; s < 16; ++s) {
      // A fragment: lane row = b0+lm, K pair = (4s + 2*half, +1) -> one b64 load
      v2f a = *(const v2f*)(xrow + 4 * s + 2 * half);
      acc0 = __builtin_amdgcn_wmma_f32_16x16x4_f32(false, a, false, bf0[s],
                                                   (short)0, acc0, false, false);
      acc1 = __builtin_amdgcn_wmma_f32_16x16x4_f32(false, a, false, bf1[s],
                                                   (short)0, acc1, false, false);
    }

    // C store: VGPR r holds row M = r + 8*half, lane lm holds col N.
    float* orow = xz + ((long)t * BB + b0) * G4H;
    #pragma unroll
    for (int r = 0; r < 8; ++r) {
      const int M = r + 8 * half;
      orow[(long)M * G4H + lm] = acc0[r];                      // lanes 0-15 / 16-31: 2x64B segments
      if (lm < G4H - 16) orow[(long)M * G4H + 16 + lm] = acc1[r];
    }
  }
}

// ---------------------------------------------------------------------------
// Kernel 2: fused 3-layer recurrence. One thread per batch element; all three
// layers advance in lockstep per timestep (layer l+1 at time t only needs
// layer l's h at time t). State = 30 regs/thread. Weights broadcast from LDS.
// ---------------------------------------------------------------------------
__device__ __forceinline__ float sigf(float x)   { return 1.0f / (1.0f + __expf(-x)); }
__device__ __forceinline__ float tanhf_(float x) { return 1.0f - 2.0f / (1.0f + __expf(2.0f * x)); }

__global__ __launch_bounds__(128) void lstm_rec(
    const float* __restrict__ xz,
    const float* __restrict__ U1,
    const float* __restrict__ W2, const float* __restrict__ U2, const float* __restrict__ b2,
    const float* __restrict__ W3, const float* __restrict__ U3, const float* __restrict__ b3,
    const float* __restrict__ Wd, const float* __restrict__ bd,
    float* __restrict__ out)
{
  __shared__ f2 sU1[50], sW2[50], sU2[50], sW3[50], sU3[50], sb2[10], sb3[10];
  __shared__ float sWd[16], sbd[4];

  const int tid = threadIdx.x;
  #define CPY(dst, src, n) { for (int i = tid; i < (n); i += blockDim.x) ((float*)(dst))[i] = (src)[i]; }
  CPY(sU1, U1, 100) CPY(sW2, W2, 100) CPY(sU2, U2, 100)
  CPY(sW3, W3, 100) CPY(sU3, U3, 100)
  CPY(sb2, b2, 20)  CPY(sb3, b3, 20)
  CPY(sWd, Wd, 15)  CPY(sbd, bd, 3)
  #undef CPY
  __syncthreads();

  const int b = blockIdx.x * blockDim.x + tid;

  float h1[HH] = {0,0,0,0,0}, c1[HH] = {0,0,0,0,0};
  float h2[HH] = {0,0,0,0,0}, c2[HH] = {0,0,0,0,0};
  float h3[HH] = {0,0,0,0,0}, c3[HH] = {0,0,0,0,0};

  for (int t = 0; t < TT; ++t) {
    const f2* zr = (const f2*)(xz + ((long)t * BB + b) * G4H);
    if (t + 1 < TT)
      __builtin_prefetch(xz + ((long)(t + 1) * BB + b) * G4H, 0, 0);  // global_prefetch_b8

    f2 z[10];
    float zf[20];

    // ---- layer 1: z = XZ[t][b] + h1 @ U1 ----
    #pragma unroll
    for (int kk = 0; kk < 10; ++kk) z[kk] = zr[kk];
    #pragma unroll
    for (int j = 0; j < HH; ++j) {
      f2 hj = { h1[j], h1[j] };
      #pragma unroll
      for (int kk = 0; kk < 10; ++kk) z[kk] += hj * sU1[j * 10 + kk];
    }
    #pragma unroll
    for (int kk = 0; kk < 10; ++kk) { zf[2*kk] = z[kk].x; zf[2*kk+1] = z[kk].y; }
    #pragma unroll
    for (int j = 0; j < HH; ++j) {
      float ig = sigf(zf[j]),        fg = sigf(zf[HH + j]);
      float gg = tanhf_(zf[2*HH+j]), og = sigf(zf[3*HH + j]);
      c1[j] = fg * c1[j] + ig * gg;
      h1[j] = og * tanhf_(c1[j]);
    }

    // ---- layer 2: z = b2 + h1 @ W2 + h2 @ U2 ----
    #pragma unroll
    for (int kk = 0; kk < 10; ++kk) z[kk] = sb2[kk];
    #pragma unroll
    for (int j = 0; j < HH; ++j) {
      f2 hj = { h1[j], h1[j] };
      #pragma unroll
      for (int kk = 0; kk < 10; ++kk) z[kk] += hj * sW2[j * 10 + kk];
    }
    #pragma unroll
    for (int j = 0; j < HH; ++j) {
      f2 hj = { h2[j], h2[j] };
      #pragma unroll
      for (int kk = 0; kk < 10; ++kk) z[kk] += hj * sU2[j * 10 + kk];
    }
    #pragma unroll
    for (int kk = 0; kk < 10; ++kk) { zf[2*kk] = z[kk].x; zf[2*kk+1] = z[kk].y; }
    #pragma unroll
    for (int j = 0; j < HH; ++j) {
      float ig = sigf(zf[j]),        fg = sigf(zf[HH + j]);
      float gg = tanhf_(zf[2*HH+j]), og = sigf(zf[3*HH + j]);
      c2[j] = fg * c2[j] + ig * gg;
      h2[j] = og * tanhf_(c2[j]);
    }

    // ---- layer 3: z = b3 + h2 @ W3 + h3 @ U3 ----
    #pragma unroll
    for (int kk = 0; kk < 10; ++kk) z[kk] = sb3[kk];
    #pragma unroll
    for (int j = 0; j < HH; ++j) {
      f2 hj = { h2[j], h2[j] };
      #pragma unroll
      for (int kk = 0; kk < 10; ++kk) z[kk] += hj * sW3[j * 10 + kk];
    }
    #pragma unroll
    for (int j = 0; j < HH; ++j) {
      f2 hj = { h3[j], h3[j] };
      #pragma unroll
      for (int kk = 0; kk < 10; ++kk) z[kk] += hj * sU3[j * 10 + kk];
    }
    #pragma unroll
    for (int kk = 0; kk < 10; ++kk) { zf[2*kk] = z[kk].x; zf[2*kk+1] = z[kk].y; }
    #pragma unroll
    for (int j = 0; j < HH; ++j) {
      float ig = sigf(zf[j]),        fg = sigf(zf[HH + j]);
      float gg = tanhf_(zf[2*HH+j]), og = sigf(zf[3*HH + j]);
      c3[j] = fg * c3[j] + ig * gg;
      h3[j] = og * tanhf_(c3[j]);
    }
  }

  // ---- dense head: out[b] = h3 @ Wd + bd ----
  #pragma unroll
  for (int o = 0; o < 3; ++o) {
    float acc = sbd[o];
    #pragma unroll
    for (int j = 0; j < HH; ++j) acc += h3[j] * sWd[j * 3 + o];
    out[(long)b * 3 + o] = acc;
  }
}

// ---------------------------------------------------------------------------
extern "C" void kernel_launch(void* const* d_in, const int* in_sizes, int n_in,
                              void* d_out, int out_size, void* d_ws, size_t ws_size,
                              hipStream_t stream)
{
  const float* x  = (const float*)d_in[0];
  const float* W1 = (const float*)d_in[1];
  const float* U1 = (const float*)d_in[2];
  const float* b1 = (const float*)d_in[3];
  const float* W2 = (const float*)d_in[4];
  const float* U2 = (const float*)d_in[5];
  const float* b2 = (const float*)d_in[6];
  const float* W3 = (const float*)d_in[7];
  const float* U3 = (const float*)d_in[8];
  const float* b3 = (const float*)d_in[9];
  const float* Wd = (const float*)d_in[10];
  const float* bd = (const float*)d_in[11];
  float* out = (float*)d_out;
  float* xzw = (float*)d_ws;   // XZ: [T][B][20] f32 = 131,072,000 bytes

  (void)in_sizes; (void)n_in; (void)out_size; (void)ws_size;

  xw1_wmma<<<1024, 256, 0, stream>>>(x, W1, b1, xzw);
  lstm_rec<<<BB / 128, 128, 0, stream>>>(xzw, U1, W2, U2, b2,
                                         W3, U3, b3, Wd, bd, out);
}